// Model_NNPZD_1D_43164421324878
// MI455X (gfx1250) — compile-verified
//
#include <hip/hip_runtime.h>
#include <stdint.h>

// ---------------- problem constants (match reference setup_inputs) ----------
#define NZ    200          // vertical cells
#define RS    264          // padded LDS row stride (floats); 264%64==8 -> tracer rows hit distinct banks
#define NH    10           // distinct Kz profiles used: h = k/4, k<40
#define SPD   8
#define LW    6            // window length (frames = LW)
#define NWIN  12
#define NSTEP ((LW - 1) * SPD)   // 40

#define C_XI   0.067f
#define C_ZETA 0.0095f
#define C_LAM  0.06f
#define C_PSI  1.46f
#define C_MU   0.25f
#define C_ETA  1.5f
#define C_DT   0.125f

typedef float v2f  __attribute__((ext_vector_type(2)));
typedef float v8f  __attribute__((ext_vector_type(8)));
typedef unsigned int u32x4 __attribute__((ext_vector_type(4)));
typedef int   i32x4 __attribute__((ext_vector_type(4)));
typedef int   i32x8 __attribute__((ext_vector_type(8)));

__global__ __launch_bounds__(32)
void nnpzd_column_kernel(const float* __restrict__ Kz,   const float* __restrict__ I0,
                         const float* __restrict__ alpha_p, const float* __restrict__ Xi_p,
                         const float* __restrict__ rho_p,   const float* __restrict__ gam_p,
                         const float* __restrict__ Gam_p,   const float* __restrict__ vphi_p,
                         const float* __restrict__ omega_p, const float* __restrict__ beta_p,
                         const float* __restrict__ N0, const float* __restrict__ A0,
                         const float* __restrict__ P0, const float* __restrict__ Z0,
                         const float* __restrict__ D0,
                         const float* __restrict__ subw,
                         const float* __restrict__ zT, const float* __restrict__ zw,
                         float* __restrict__ out,
                         int KzD1, int KzD2, int I0D1)
{
  // ---------------- LDS (all state resident; ~48 KB / workgroup) ------------
  __shared__ float st[5][RS];           // tracers N, NH4, P, Z, D (rhs/solution in place)
  __shared__ float cum[RS];             // cumsum scratch (also reused as sediment flux[0..NZ])
  __shared__ float ainf_s[NH][RS];      // sub-diagonal a_i per h
  __shared__ float cp_s[NH][RS];        // Thomas c' per h
  __shared__ float im_s[NH][RS];        // Thomas 1/m per h
  __shared__ float dzT_s[RS];
  __shared__ float dzw_s[RS + 8];
  __shared__ float expz_s[RS];          // exp(-XI * zT)
  __shared__ float kzbuf[NH * 201 + 8]; // TDM destination: 10 x (Nz+1) Kz rows
  __shared__ float rowsum[16];

  const int l   = threadIdx.x;          // 0..31, one wave
  const int blk = blockIdx.x;
  const int s   = blk / NWIN;
  const int w   = blk % NWIN;
  const int sw0 = (int)subw[w * LW + 0];

  // ---------------- geometry ------------------------------------------------
  for (int z = l; z < NZ; z += 32) {
    dzT_s[z]  = zw[z + 1] - zw[z];
    expz_s[z] = __expf(-C_XI * zT[z]);
  }
  for (int z = l; z <= NZ; z += 32) {
    float d;
    if (z == 0)       d = 0.5f * (zw[1] - zw[0]);
    else if (z == NZ) d = 0.5f * (zw[NZ] - zw[NZ - 1]);
    else              d = zT[z] - zT[z - 1];
    dzw_s[z] = d;
  }

  // ---------------- TDM: async tensor load of Kz window (10 x 201 f32) ------
  {
    // low 32 bits of a flat LDS pointer are the LDS byte offset (ISA 10.2 aperture rule)
    uint32_t lds_off = (uint32_t)(uintptr_t)&kzbuf[0];
    uint64_t ga = (uint64_t)(uintptr_t)(Kz + ((size_t)s * KzD1 + 2 * sw0) * KzD2);
    uint32_t td0 = (uint32_t)KzD2;              // tensor dim0 (row length)
    uint32_t td1 = (uint32_t)(KzD1 - 2 * sw0);  // rows available from tile start
    uint32_t tl0 = (uint32_t)KzD2, tl1 = NH, tl2 = 0;
    uint64_t st0 = (uint64_t)KzD2;              // dim0 stride (elements)
    u32x4 g0; i32x8 g1; i32x4 g2, g3; i32x8 g4;
    g0.x = 1u;                                  // count = 1 valid descriptor
    g0.y = lds_off;                             // lds_addr [63:32]
    g0.z = (uint32_t)ga;                        // global_addr [95:64]
    g0.w = (uint32_t)((ga >> 32) & 0x01FFFFFFu) | (2u << 30);  // addr hi + type=2
    g1[0] = (int)(2u << 16);                                   // data_size = 4B
    g1[1] = (int)((td0 & 0xFFFFu) << 16);                      // tensor_dim0 lo
    g1[2] = (int)(((td0 >> 16) & 0xFFFFu) | ((td1 & 0xFFFFu) << 16));
    g1[3] = (int)(((td1 >> 16) & 0xFFFFu) | ((tl0 & 0xFFFFu) << 16));
    g1[4] = (int)((tl1 & 0xFFFFu) | ((tl2 & 0xFFFFu) << 16));
    g1[5] = (int)(uint32_t)(st0 & 0xFFFFFFFFu);                // dim0 stride lo32
    g1[6] = (int)((uint32_t)((st0 >> 32) & 0xFFFFu));          // stride hi, dim1 stride=0
    g1[7] = 0;
    g2[0] = g2[1] = g2[2] = g2[3] = 0;
    g3[0] = g3[1] = g3[2] = g3[3] = 0;
    g4[0] = g4[1] = g4[2] = g4[3] = g4[4] = g4[5] = g4[6] = g4[7] = 0;
    __builtin_amdgcn_tensor_load_to_lds(g0, g1, g2, g3, g4, 0);
  }

  // ---------------- initial state load + frame 0 ----------------------------
  const size_t coloff = ((size_t)s * NWIN + w) * NZ;
  float* outcol = out + ((size_t)s * NWIN + w) * ((size_t)LW * 5 * NZ);
  for (int z = l; z < NZ; z += 32) {
    float vN = N0[coloff + z], vA = A0[coloff + z], vP = P0[coloff + z],
          vZ = Z0[coloff + z], vD = D0[coloff + z];
    st[0][z] = vN; st[1][z] = vA; st[2][z] = vP; st[3][z] = vZ; st[4][z] = vD;
    outcol[0 * NZ + z] = vN; outcol[1 * NZ + z] = vA; outcol[2 * NZ + z] = vP;
    outcol[3 * NZ + z] = vZ; outcol[4 * NZ + z] = vD;
  }

  __builtin_amdgcn_s_wait_tensorcnt(0);
  __syncthreads();

  // ---------------- factorize the 10 tridiagonal matrices (10 lanes, once) --
  if (l < NH) {
    const float* K  = &kzbuf[l * KzD2];
    float* As = ainf_s[l]; float* Cp = cp_s[l]; float* Im = im_s[l];
    float sup = -C_DT * K[1] / (dzT_s[0] * dzw_s[1]);
    float b   = 1.0f - sup;
    float im  = 1.0f / b;
    float cpp = sup * im;
    As[0] = 0.0f; Im[0] = im; Cp[0] = cpp;
    for (int i = 1; i < NZ; ++i) {
      float sub  = -C_DT * K[i] / (dzT_s[i] * dzw_s[i]);
      float sup2 = (i < NZ - 1) ? -C_DT * K[i + 1] / (dzT_s[i] * dzw_s[i + 1]) : 0.0f;
      float bb   = 1.0f - sub - sup2;
      float m    = bb - sub * cpp;
      float imi  = 1.0f / m;
      float cpi  = sup2 * imi;
      As[i] = sub; Im[i] = imi; Cp[i] = cpi;
      cpp = cpi;
    }
  }
  __syncthreads();

  const float a_c   = alpha_p[s], Xi_c = Xi_p[s], rho_c = rho_p[s], gam_c = gam_p[s],
              Gam_c = Gam_p[s], vphi_c = vphi_p[s], sed_c = omega_p[s], bet_c = beta_p[s];

  const bool hi = (l >= 16);
  const int  m  = l & 15;

  // ---------------- time stepping -------------------------------------------
  for (int k = 0; k < NSTEP; ++k) {
    const int d = k >> 3;
    const int h = k >> 2;
    const float I0d = I0[(size_t)s * I0D1 + sw0 + d];

    // (1) v = P * dzT, zero-padded to 256
    for (int z = l; z < 256; z += 32)
      cum[z] = (z < NZ) ? st[2][z] * dzT_s[z] : 0.0f;
    __syncthreads();

    // (2) inclusive 256-prefix on the tensor pipe:
    //     S = M*U + Lstrict*(M*J)   (M: 16x16 tile, row i = v[16i..16i+15])
    {
      const int o = hi ? 2 : 0;
      v2f a0, a1, a2, a3;
      a0.x = cum[16 * m + 0  + o]; a0.y = cum[16 * m + 1  + o];
      a1.x = cum[16 * m + 4  + o]; a1.y = cum[16 * m + 5  + o];
      a2.x = cum[16 * m + 8  + o]; a2.y = cum[16 * m + 9  + o];
      a3.x = cum[16 * m + 12 + o]; a3.y = cum[16 * m + 13 + o];

      v2f ones; ones.x = 1.0f; ones.y = 1.0f;
      v8f t = {0.f, 0.f, 0.f, 0.f, 0.f, 0.f, 0.f, 0.f};
      t = __builtin_amdgcn_wmma_f32_16x16x4_f32(false, a0, false, ones, (short)0, t, false, false);
      t = __builtin_amdgcn_wmma_f32_16x16x4_f32(false, a1, false, ones, (short)0, t, false, false);
      t = __builtin_amdgcn_wmma_f32_16x16x4_f32(false, a2, false, ones, (short)0, t, false, false);
      t = __builtin_amdgcn_wmma_f32_16x16x4_f32(false, a3, false, ones, (short)0, t, false, false);
      if (l == 0)  { rowsum[0] = t[0]; rowsum[1] = t[1]; rowsum[2] = t[2]; rowsum[3] = t[3];
                     rowsum[4] = t[4]; rowsum[5] = t[5]; rowsum[6] = t[6]; rowsum[7] = t[7]; }
      if (l == 16) { rowsum[8] = t[0]; rowsum[9] = t[1]; rowsum[10] = t[2]; rowsum[11] = t[3];
                     rowsum[12] = t[4]; rowsum[13] = t[5]; rowsum[14] = t[6]; rowsum[15] = t[7]; }
      __syncthreads();

      v8f acc = {0.f, 0.f, 0.f, 0.f, 0.f, 0.f, 0.f, 0.f};
#pragma unroll
      for (int c = 0; c < 4; ++c) {                 // acc += M_c * U_c
        const int k0 = 4 * c + (hi ? 2 : 0), k1 = 4 * c + (hi ? 3 : 1);
        v2f bu; bu.x = (k0 <= m) ? 1.0f : 0.0f; bu.y = (k1 <= m) ? 1.0f : 0.0f;
        v2f ac = (c == 0) ? a0 : (c == 1) ? a1 : (c == 2) ? a2 : a3;
        acc = __builtin_amdgcn_wmma_f32_16x16x4_f32(false, ac, false, bu, (short)0, acc, false, false);
      }
#pragma unroll
      for (int c = 0; c < 4; ++c) {                 // acc += Lstrict_c * T_c (T bcast rowsums)
        const int k0 = 4 * c + (hi ? 2 : 0), k1 = 4 * c + (hi ? 3 : 1);
        v2f al; al.x = (k0 < m) ? 1.0f : 0.0f; al.y = (k1 < m) ? 1.0f : 0.0f;
        v2f bt; bt.x = rowsum[k0]; bt.y = rowsum[k1];
        acc = __builtin_amdgcn_wmma_f32_16x16x4_f32(false, al, false, bt, (short)0, acc, false, false);
      }
#pragma unroll
      for (int r = 0; r < 8; ++r) {                 // D layout -> cum[]
        const int row = hi ? (r + 8) : r;
        cum[16 * row + m] = acc[r];
      }
    }
    __syncthreads();

    // (3) biology -> rhs in place
    for (int z = l; z < NZ; z += 32) {
      float N  = st[0][z], A4 = st[1][z], P = st[2][z], Zp = st[3][z], Dd = st[4][z];
      float It = I0d * expz_s[z] * __expf(-C_ZETA * cum[z]);
      float aI = a_c * It;
      float G  = C_ETA * aI * rsqrtf(C_ETA * C_ETA + aI * aI);
      float fN = N  / (1.0f + N);
      float fA = A4 / (1.0f + A4);
      float eP = __expf(-C_PSI * A4);
      float gr = rho_c * (1.0f - __expf(-C_LAM * P)) * Zp;
      float bN = C_MU * A4 - G * eP * P * fN;
      float bA = vphi_c * Dd + Gam_c * Zp - G * P * fA - C_MU * A4;
      float bP = G * P * (eP * fN + fA) - gr - Xi_c * P;
      float bZ = (1.0f - gam_c) * gr - Gam_c * Zp;
      float bD = gam_c * gr + Xi_c * P - vphi_c * Dd;
      st[0][z] = N  + C_DT * bN;
      st[1][z] = A4 + C_DT * bA;
      st[2][z] = P  + C_DT * bP;
      st[3][z] = Zp + C_DT * bZ;
      st[4][z] = Dd + C_DT * bD;
    }
    __syncthreads();

    // (4) Thomas solve: 5 tracers on 5 lanes, shared precomputed factorization
    if (l < 5) {
      float* dd = st[l];
      const float* As = ainf_s[h];
      const float* Cp = cp_s[h];
      const float* Im = im_s[h];
      float dp = dd[0] * Im[0];
      dd[0] = dp;
      for (int i = 1; i < NZ; ++i) { dp = (dd[i] - As[i] * dp) * Im[i]; dd[i] = dp; }
      float x = dp;
      for (int i = NZ - 2; i >= 0; --i) { x = dd[i] - Cp[i] * x; dd[i] = x; }
    }
    __syncthreads();

    // (5) WENO3-Z sediment flux on D (cum[] reused as flux[0..NZ])
    for (int z = l; z <= NZ; z += 32) {
      float F;
      if (z == 0)       F = 0.0f;
      else if (z == 1)  F = sed_c * st[4][0];
      else if (z == NZ) F = sed_c * st[4][NZ - 1];
      else {
        float qm = st[4][z - 2] * dzT_s[z - 2];
        float q0 = st[4][z - 1] * dzT_s[z - 1];
        float qp = st[4][z]     * dzT_s[z];
        float qi1 = -0.5f * qm + 1.5f * q0;
        float qi2 = 0.5f * (q0 + qp);
        float b1 = (q0 - qm) * (q0 - qm);
        float b2 = (qp - q0) * (qp - q0);
        float tau = fabsf(b2 - b1);
        float w1 = (1.0f / 3.0f) * (1.0f + tau / (b1 + 1e-14f));
        float w2 = (2.0f / 3.0f) * (1.0f + tau / (b2 + 1e-14f));
        F = sed_c * ((w1 * qi1 + w2 * qi2) / (w1 + w2)) / dzw_s[z];
      }
      cum[z] = F;
    }
    __syncthreads();

    // (6) sediment update + river relaxation
    {
      float vs  = subw[w * LW + (d + 1)];
      float rap = 0.05f * (fmaxf(vs - 30.f, 0.f) - fmaxf(vs - 31.f, 0.f)
                         - fmaxf(vs - 60.f, 0.f) + fmaxf(vs - 61.f, 0.f));
      for (int z = l; z < NZ; z += 32) {
        st[4][z] += C_DT * (cum[z] - cum[z + 1]) / dzT_s[z];
        st[0][z] += (bet_c - st[0][z]) * rap;
      }
    }
    __syncthreads();

    // (7) snapshot every SPD steps
    if ((k & (SPD - 1)) == (SPD - 1)) {
      const int f = (k + 1) >> 3;
      for (int tr = 0; tr < 5; ++tr)
        for (int z = l; z < NZ; z += 32)
          outcol[((size_t)f * 5 + tr) * NZ + z] = st[tr][z];
    }
  }
}

extern "C" void kernel_launch(void* const* d_in, const int* in_sizes, int n_in,
                              void* d_out, int out_size, void* d_ws, size_t ws_size,
                              hipStream_t stream) {
  (void)n_in; (void)out_size; (void)d_ws; (void)ws_size;
  const float* Kz    = (const float*)d_in[0];
  const float* I0    = (const float*)d_in[1];
  const float* alpha = (const float*)d_in[2];
  const float* Xi    = (const float*)d_in[3];
  const float* rho   = (const float*)d_in[4];
  const float* gam   = (const float*)d_in[5];
  const float* Gam   = (const float*)d_in[6];
  const float* vphi  = (const float*)d_in[7];
  const float* omeg  = (const float*)d_in[8];
  const float* beta  = (const float*)d_in[9];
  const float* N0    = (const float*)d_in[10];
  const float* A0    = (const float*)d_in[11];
  const float* P0    = (const float*)d_in[12];
  const float* Z0    = (const float*)d_in[13];
  const float* D0    = (const float*)d_in[14];
  const float* subw  = (const float*)d_in[15];
  const float* zT    = (const float*)d_in[16];
  const float* zw    = (const float*)d_in[17];
  float* out = (float*)d_out;

  const int Ns   = in_sizes[2];
  const int Nz   = in_sizes[16];
  const int KzD2 = Nz + 1;
  const int KzD1 = in_sizes[0] / (Ns * KzD2);
  const int I0D1 = in_sizes[1] / Ns;
  const int nwv  = in_sizes[10] / (Ns * Nz);

  dim3 grid(Ns * nwv), block(32);
  nnpzd_column_kernel<<<grid, block, 0, stream>>>(
      Kz, I0, alpha, Xi, rho, gam, Gam, vphi, omeg, beta,
      N0, A0, P0, Z0, D0, subw, zT, zw, out, KzD1, KzD2, I0D1);
}